// MaskedMultiheadSelfAttention_Alibi_8297876816408
// MI455X (gfx1250) — compile-verified
//
#include <hip/hip_runtime.h>
#include <hip/hip_bf16.h>
#include <math.h>

// ---------------------------------------------------------------------------
// MaskedMultiheadSelfAttention + ALiBi, fp32, CDNA5 (gfx1250, wave32)
// Uses V_WMMA_F32_16X16X4_F32 for all matmuls.
// N=2, S=2048, E=1024, H=16, D=64
// ---------------------------------------------------------------------------

typedef float v2f __attribute__((ext_vector_type(2)));
typedef float v8f __attribute__((ext_vector_type(8)));

static constexpr int N_ = 2;
static constexpr int S_ = 2048;
static constexpr int E_ = 1024;
static constexpr int H_ = 16;
static constexpr int D_ = 64;

static __device__ __forceinline__ v8f wmma_f32(v2f a, v2f b, v8f c) {
  // D = A(16x4) * B(4x16) + C(16x16), fp32
  return __builtin_amdgcn_wmma_f32_16x16x4_f32(false, a, false, b,
                                               (short)0, c, false, false);
}

static __device__ __forceinline__ float rowmax16(float v) {
  v = fmaxf(v, __shfl_xor(v, 1, 32));
  v = fmaxf(v, __shfl_xor(v, 2, 32));
  v = fmaxf(v, __shfl_xor(v, 4, 32));
  v = fmaxf(v, __shfl_xor(v, 8, 32));
  return v;
}
static __device__ __forceinline__ float rowsum16(float v) {
  v += __shfl_xor(v, 1, 32);
  v += __shfl_xor(v, 2, 32);
  v += __shfl_xor(v, 4, 32);
  v += __shfl_xor(v, 8, 32);
  return v;
}

// ---------------------------------------------------------------------------
// Kernel 1: per-head projection.  Out[n,h,s,:] = X[n,s,h*64:(h+1)*64] @ W^T
// grid = (S/64, H, N), block = 128 (4 waves; each wave: 16 rows x 64 cols)
// ---------------------------------------------------------------------------
__global__ __launch_bounds__(128)
void proj_kernel(const float* __restrict__ X,   // [N,S,E]
                 const float* __restrict__ W,   // [D,D] row-major, Out = X W^T
                 float* __restrict__ Out)       // [N,H,S,D]
{
  __shared__ float Ws[64 * 68];                 // padded stride 68
  const int tid  = threadIdx.x;
  const int wave = tid >> 5;
  const int lane = tid & 31;
  const int n  = blockIdx.z, h = blockIdx.y;
  const int s0 = blockIdx.x * 64 + wave * 16;
  const int mrow = lane & 15, half = lane >> 4;

  // stage W (64x64) into LDS
  for (int i = tid; i < 1024; i += 128) {
    const int r = i >> 4, c4 = i & 15;
    const float4 w4 = ((const float4*)W)[r * 16 + c4];
    float* d = &Ws[r * 68 + c4 * 4];
    d[0] = w4.x; d[1] = w4.y; d[2] = w4.z; d[3] = w4.w;
  }
  __syncthreads();

  const float* xrow = X + ((size_t)n * S_ + (s0 + mrow)) * E_ + h * D_;
  v8f c[4] = {};
  #pragma unroll
  for (int ks = 0; ks < 16; ++ks) {
    const int k = 4 * ks + 2 * half;            // A: k = 2*(lane/16)+vgpr
    v2f a; a.x = xrow[k]; a.y = xrow[k + 1];
    #pragma unroll
    for (int f = 0; f < 4; ++f) {
      const int e = 16 * f + mrow;              // B: n = lane%16
      v2f b; b.x = Ws[e * 68 + k]; b.y = Ws[e * 68 + k + 1];
      c[f] = wmma_f32(a, b, c[f]);
    }
  }

  float* obase = Out + ((size_t)(n * H_ + h)) * S_ * D_;
  #pragma unroll
  for (int f = 0; f < 4; ++f)
    #pragma unroll
    for (int r = 0; r < 8; ++r) {
      const int m = r + 8 * half;               // C: m = vgpr + 8*(lane/16)
      const int e = 16 * f + mrow;              // C: n = lane%16
      obase[(size_t)(s0 + m) * D_ + e] = c[f][r];
    }
}

// ---------------------------------------------------------------------------
// Kernel 2: fused causal flash attention with ALiBi.
// grid = (S/64, H, N), block = 128. Each wave: 16 queries x full D=64.
// softmax input = (q.k + slope*(k-q)) / sqrt(E), masked where k > q.
// ---------------------------------------------------------------------------
__global__ __launch_bounds__(128)
void flash_kernel(const float* __restrict__ Qp,   // [N,H,S,D]
                  const float* __restrict__ Kp,   // [N,H,S,D]
                  const float* __restrict__ Vp,   // [N,H,S,D]
                  float* __restrict__ O)          // [N,S,E] (head-interleaved)
{
  __shared__ float Ks[16 * 68];
  __shared__ float Vs[16 * 68];
  __shared__ float Ps[4][16 * 17];

  const int tid  = threadIdx.x;
  const int wave = tid >> 5;
  const int lane = tid & 31;
  const int n = blockIdx.z, h = blockIdx.y;
  const int qblk = blockIdx.x;
  const int q0 = qblk * 64 + wave * 16;
  const int mrow = lane & 15, half = lane >> 4;
  const size_t headOff = ((size_t)(n * H_ + h)) * S_ * D_;

  // preload this wave's Q tile into A-fragment registers (16 k-steps x v2f)
  float qreg[32];
  const float* qrow = Qp + headOff + (size_t)(q0 + mrow) * D_;
  #pragma unroll
  for (int ks = 0; ks < 16; ++ks) {
    const int k = 4 * ks + 2 * half;
    qreg[2 * ks]     = qrow[k];
    qreg[2 * ks + 1] = qrow[k + 1];
  }

  v8f o[4] = {};
  float mrun[8], lrun[8];
  #pragma unroll
  for (int r = 0; r < 8; ++r) { mrun[r] = -3.0e38f; lrun[r] = 0.0f; }

  const float slope  = (float)(1 << __popc(h));  // reference slopes: 2^popcount(h)
  const float scale  = 0.03125f;                 // 1/sqrt(E) = 1/32
  const int   ktMax  = (qblk * 64 + 63) >> 4;    // last key tile for this block

  for (int kt = 0; kt <= ktMax; ++kt) {
    const int k0 = kt * 16;
    __syncthreads();
    // cooperative K/V tile load (16x64 each) -> padded LDS
    for (int i = tid; i < 256; i += 128) {
      const int r = i >> 4, c4 = i & 15;
      const float4 kq = ((const float4*)(Kp + headOff + (size_t)(k0 + r) * D_))[c4];
      const float4 vq = ((const float4*)(Vp + headOff + (size_t)(k0 + r) * D_))[c4];
      float* kd = &Ks[r * 68 + c4 * 4];
      float* vd = &Vs[r * 68 + c4 * 4];
      kd[0] = kq.x; kd[1] = kq.y; kd[2] = kq.z; kd[3] = kq.w;
      vd[0] = vq.x; vd[1] = vq.y; vd[2] = vq.z; vd[3] = vq.w;
    }
    __syncthreads();

    const bool active = (k0 <= q0 + 15);         // wave-uniform -> EXEC all-1s
    if (active) {
      // S-tile = Q(16x64) x K^T(64x16)
      v8f s = {};
      #pragma unroll
      for (int ks = 0; ks < 16; ++ks) {
        const int k = 4 * ks + 2 * half;
        v2f a; a.x = qreg[2 * ks]; a.y = qreg[2 * ks + 1];
        v2f b; b.x = Ks[mrow * 68 + k]; b.y = Ks[mrow * 68 + k + 1];
        s = wmma_f32(a, b, s);
      }
      // ALiBi + causal mask + 1/sqrt(E)
      #pragma unroll
      for (int r = 0; r < 8; ++r) {
        const int q = q0 + r + 8 * half;
        const int kcol = k0 + mrow;
        float e = (s[r] + slope * (float)(kcol - q)) * scale;
        if (kcol > q) e = -3.0e18f;              // -1e20/32 (masked)
        s[r] = e;
      }
      // online softmax update
      float pscale[8];
      #pragma unroll
      for (int r = 0; r < 8; ++r) {
        const float rm   = rowmax16(s[r]);
        const float mnew = fmaxf(mrun[r], rm);
        const float alpha = __expf(mrun[r] - mnew);
        mrun[r] = mnew;
        const float p = __expf(s[r] - mnew);
        s[r] = p;
        lrun[r] = lrun[r] * alpha + rowsum16(p);
        pscale[r] = alpha;
      }
      #pragma unroll
      for (int f = 0; f < 4; ++f)
        #pragma unroll
        for (int r = 0; r < 8; ++r) o[f][r] *= pscale[r];

      // re-layout P (C-frag) -> LDS -> A-frag, then O += P(16x16) x V(16x64)
      #pragma unroll
      for (int r = 0; r < 8; ++r)
        Ps[wave][(r + 8 * half) * 17 + mrow] = s[r];
      #pragma unroll
      for (int kk = 0; kk < 4; ++kk) {
        const int ka = 4 * kk + 2 * half;
        v2f a; a.x = Ps[wave][mrow * 17 + ka]; a.y = Ps[wave][mrow * 17 + ka + 1];
        #pragma unroll
        for (int f = 0; f < 4; ++f) {
          const int dcol = 16 * f + mrow;
          v2f b;
          b.x = Vs[(4 * kk + 2 * half)     * 68 + dcol];
          b.y = Vs[(4 * kk + 2 * half + 1) * 68 + dcol];
          o[f] = wmma_f32(a, b, o[f]);
        }
      }
    }
  }

  // normalize and write [N,S,E] with head offset
  #pragma unroll
  for (int r = 0; r < 8; ++r) lrun[r] = 1.0f / lrun[r];
  float* obase = O + (size_t)n * S_ * E_ + h * D_;
  #pragma unroll
  for (int f = 0; f < 4; ++f)
    #pragma unroll
    for (int r = 0; r < 8; ++r) {
      const int m = r + 8 * half;
      const int dcol = 16 * f + mrow;
      obase[(size_t)(q0 + m) * E_ + dcol] = o[f][r] * lrun[r];
    }
}

// ---------------------------------------------------------------------------
// Kernel 3: output projection.  Y = A @ Wo^T + bo,  A:[4096,1024], Wo:[1024,1024]
// grid = (E/64, R/64), block = 128 (4 waves x 16 rows each)
// ---------------------------------------------------------------------------
__global__ __launch_bounds__(128)
void outproj_kernel(const float* __restrict__ A,
                    const float* __restrict__ Wo,
                    const float* __restrict__ bo,
                    float* __restrict__ Y)
{
  __shared__ float Ws[64 * 68];
  const int tid  = threadIdx.x;
  const int wave = tid >> 5;
  const int lane = tid & 31;
  const int r0 = blockIdx.y * 64 + wave * 16;
  const int c0 = blockIdx.x * 64;
  const int mrow = lane & 15, half = lane >> 4;

  v8f c[4] = {};
  const float* arow = A + (size_t)(r0 + mrow) * E_;

  for (int kc = 0; kc < E_; kc += 64) {
    __syncthreads();
    // stage Wo[c0:c0+64, kc:kc+64]
    for (int i = tid; i < 1024; i += 128) {
      const int rr = i >> 4, c4 = i & 15;
      const float4 w4 = ((const float4*)(Wo + (size_t)(c0 + rr) * E_ + kc))[c4];
      float* d = &Ws[rr * 68 + c4 * 4];
      d[0] = w4.x; d[1] = w4.y; d[2] = w4.z; d[3] = w4.w;
    }
    __syncthreads();
    #pragma unroll
    for (int ks = 0; ks < 16; ++ks) {
      const int k = 4 * ks + 2 * half;
      v2f a; a.x = arow[kc + k]; a.y = arow[kc + k + 1];
      #pragma unroll
      for (int f = 0; f < 4; ++f) {
        const int e = 16 * f + mrow;
        v2f b; b.x = Ws[e * 68 + k]; b.y = Ws[e * 68 + k + 1];
        c[f] = wmma_f32(a, b, c[f]);
      }
    }
  }

  #pragma unroll
  for (int f = 0; f < 4; ++f)
    #pragma unroll
    for (int r = 0; r < 8; ++r) {
      const int m = r + 8 * half;
      const int e = c0 + 16 * f + mrow;
      Y[(size_t)(r0 + m) * E_ + e] = c[f][r] + bo[e];
    }
}

// ---------------------------------------------------------------------------
// Launch.  Inputs: 0=values 1=keys 2=query 3=mask 4=Wv 5=Wk 6=Wq 7=Wo 8=bo
// Workspace: Qp|Kp|Vp|Oattn, each N*H*S*D = 4,194,304 floats (16 MB) -> 64 MB
// ---------------------------------------------------------------------------
extern "C" void kernel_launch(void* const* d_in, const int* in_sizes, int n_in,
                              void* d_out, int out_size, void* d_ws, size_t ws_size,
                              hipStream_t stream) {
  const float* values = (const float*)d_in[0];
  const float* keys   = (const float*)d_in[1];
  const float* query  = (const float*)d_in[2];
  const float* Wv     = (const float*)d_in[4];
  const float* Wk     = (const float*)d_in[5];
  const float* Wq     = (const float*)d_in[6];
  const float* Wo     = (const float*)d_in[7];
  const float* bo     = (const float*)d_in[8];
  float* out = (float*)d_out;

  const size_t tensorElems = (size_t)N_ * H_ * S_ * D_;   // 4,194,304
  float* Qp = (float*)d_ws;
  float* Kp = Qp + tensorElems;
  float* Vp = Kp + tensorElems;
  float* Oa = Vp + tensorElems;

  const dim3 blk(128);
  const dim3 gProj(S_ / 64, H_, N_);

  proj_kernel<<<gProj, blk, 0, stream>>>(query,  Wq, Qp);
  proj_kernel<<<gProj, blk, 0, stream>>>(keys,   Wk, Kp);
  proj_kernel<<<gProj, blk, 0, stream>>>(values, Wv, Vp);

  flash_kernel<<<gProj, blk, 0, stream>>>(Qp, Kp, Vp, Oa);

  const dim3 gOut(E_ / 64, (N_ * S_) / 64);
  outproj_kernel<<<gOut, blk, 0, stream>>>(Oa, Wo, bo, out);
}